// TMC_8057358647586
// MI455X (gfx1250) — compile-verified
//
#include <hip/hip_runtime.h>

typedef __attribute__((ext_vector_type(2))) float v2f;
typedef __attribute__((ext_vector_type(8))) float v8f;

#define B_ROWS 131072
#define KCL 512
#define DIM 64

// ---------- kernel 1: c_sq[k] = sum_d C[k][d]^2 ----------
__global__ void TMC_csq_kernel(const float* __restrict__ C, float* __restrict__ csq) {
  int k = blockIdx.x * blockDim.x + threadIdx.x;
  if (k < KCL) {
    const float4* row = (const float4*)(C + (long)k * DIM);
    float s = 0.f;
#pragma unroll
    for (int i = 0; i < DIM / 4; ++i) {
      float4 v = row[i];
      s += v.x * v.x + v.y * v.y + v.z * v.z + v.w * v.w;
    }
    csq[k] = s;
  }
}

// ---------- kernel 2: main. 16 rows x 512 cols per workgroup ----------
__global__ __launch_bounds__(256) void TMC_cluster_kernel(
    const float* __restrict__ z, const float* __restrict__ C,
    const float* __restrict__ csq, float* __restrict__ q,
    float* __restrict__ partial) {
  __shared__ float pz[16][17];
  __shared__ float zsq[16];
  __shared__ float wred[8][16];
  __shared__ float wsumS[8][16];
  __shared__ float wdotS[8][16];
  __shared__ float rowmaxS[16];
  __shared__ float rsinvS[16];
  __shared__ float rlossS[16];

  const int tid  = threadIdx.x;
  const int wave = tid >> 5;
  const int lane = tid & 31;
  const int n    = lane & 15;   // column within 16-wide tile / row M for A-frag
  const int hi   = lane >> 4;   // K-pair selector (0 -> k,k+1 ; 1 -> k+2,k+3)
  const long m0  = (long)blockIdx.x * 16;

  // ---- z_sq for the 16 rows (cooperative, deterministic) ----
  {
    int row = tid >> 4, part = tid & 15;
    const float4 v = *(const float4*)(z + (m0 + row) * DIM + part * 4);
    pz[row][part] = v.x * v.x + v.y * v.y + v.z * v.z + v.w * v.w;
  }
  __syncthreads();
  if (tid < 16) {
    float s = 0.f;
#pragma unroll
    for (int i = 0; i < 16; ++i) s += pz[tid][i];
    zsq[tid] = s;
  }
  __syncthreads();

  // ---- A fragments: 16x4 f32 tiles, 16 chunks covering K=64 ----
  // lane<16 holds (M=lane, K = 4kc+{0,1}); lane>=16 holds (M=lane-16, K = 4kc+{2,3})
  v2f a[16];
  const float* zrow = z + (m0 + n) * DIM + 2 * hi;
#pragma unroll
  for (int kc = 0; kc < 16; ++kc) a[kc] = *(const v2f*)(zrow + kc * 4);

  float myzsq[8];
#pragma unroll
  for (int r = 0; r < 8; ++r) myzsq[r] = zsq[r + 8 * hi];

  // ---- cross GEMM via V_WMMA_F32_16X16X4_F32; s = -dist_sq ----
  float s[4][8];
#pragma unroll
  for (int t = 0; t < 4; ++t) {
    const int n0 = wave * 64 + t * 16;
    v8f acc = {};
    const float* crow = C + (long)(n0 + n) * DIM + 2 * hi;
#pragma unroll
    for (int kc = 0; kc < 16; ++kc) {
      v2f b = *(const v2f*)(crow + kc * 4);
      acc = __builtin_amdgcn_wmma_f32_16x16x4_f32(
          false, a[kc], false, b, (short)0, acc, false, false);
    }
    const float cs = csq[n0 + n];
#pragma unroll
    for (int r = 0; r < 8; ++r)
      s[t][r] = 2.0f * acc[r] - myzsq[r] - cs;  // = -dist_sq  (tau == 1)
  }

  // ---- row max (over 512 cols): lane-tree -> per-wave LDS -> fixed-order ----
  float rp[8];
#pragma unroll
  for (int r = 0; r < 8; ++r)
    rp[r] = fmaxf(fmaxf(s[0][r], s[1][r]), fmaxf(s[2][r], s[3][r]));
#pragma unroll
  for (int off = 8; off >= 1; off >>= 1)
#pragma unroll
    for (int r = 0; r < 8; ++r) rp[r] = fmaxf(rp[r], __shfl_xor(rp[r], off, 32));
  if (n == 0)
#pragma unroll
    for (int r = 0; r < 8; ++r) wred[wave][r + 8 * hi] = rp[r];
  __syncthreads();
  if (tid < 16) {
    float m = wred[0][tid];
#pragma unroll
    for (int wv = 1; wv < 8; ++wv) m = fmaxf(m, wred[wv][tid]);
    rowmaxS[tid] = m;
  }
  __syncthreads();

  // ---- exp, row sum, and row dot (for loss) ----
  float rmax[8];
#pragma unroll
  for (int r = 0; r < 8; ++r) rmax[r] = rowmaxS[r + 8 * hi];
  float psum[8], pdot[8];
#pragma unroll
  for (int r = 0; r < 8; ++r) { psum[r] = 0.f; pdot[r] = 0.f; }
#pragma unroll
  for (int t = 0; t < 4; ++t)
#pragma unroll
    for (int r = 0; r < 8; ++r) {
      float sv = s[t][r];
      float e  = __expf(sv - rmax[r]);
      psum[r] += e;
      pdot[r] += e * (-sv);   // e * dist_sq
      s[t][r] = e;
    }
#pragma unroll
  for (int off = 8; off >= 1; off >>= 1)
#pragma unroll
    for (int r = 0; r < 8; ++r) {
      psum[r] += __shfl_xor(psum[r], off, 32);
      pdot[r] += __shfl_xor(pdot[r], off, 32);
    }
  if (n == 0)
#pragma unroll
    for (int r = 0; r < 8; ++r) {
      wsumS[wave][r + 8 * hi] = psum[r];
      wdotS[wave][r + 8 * hi] = pdot[r];
    }
  __syncthreads();
  if (tid < 16) {
    float ssum = 0.f, sdot = 0.f;
#pragma unroll
    for (int wv = 0; wv < 8; ++wv) { ssum += wsumS[wv][tid]; sdot += wdotS[wv][tid]; }
    rsinvS[tid] = 1.0f / ssum;
    rlossS[tid] = sdot / ssum;   // sum_k q*dist for this row
  }
  __syncthreads();
  if (tid == 0) {
    float p = 0.f;
#pragma unroll
    for (int i = 0; i < 16; ++i) p += rlossS[i];
    partial[blockIdx.x] = p;     // overwrite: deterministic across replays
  }

  // ---- write q (streaming, non-temporal: 268 MB written once) ----
  float ri[8];
#pragma unroll
  for (int r = 0; r < 8; ++r) ri[r] = rsinvS[r + 8 * hi];
#pragma unroll
  for (int t = 0; t < 4; ++t) {
    const int n0 = wave * 64 + t * 16;
#pragma unroll
    for (int r = 0; r < 8; ++r) {
      long row = m0 + r + 8 * hi;
      __builtin_nontemporal_store(s[t][r] * ri[r], q + row * KCL + n0 + n);
    }
  }
}

// ---------- kernel 3: deterministic fixed-order loss reduction ----------
__global__ void TMC_loss_kernel(const float* __restrict__ partial,
                                float* __restrict__ out, int nblocks) {
  __shared__ float buf[256];
  float s = 0.f;
  for (int i = threadIdx.x; i < nblocks; i += 256) s += partial[i];
  buf[threadIdx.x] = s;
  __syncthreads();
  for (int off = 128; off >= 1; off >>= 1) {
    if ((int)threadIdx.x < off) buf[threadIdx.x] += buf[threadIdx.x + off];
    __syncthreads();
  }
  if (threadIdx.x == 0) out[0] = buf[0] / (float)B_ROWS;
}

extern "C" void kernel_launch(void* const* d_in, const int* in_sizes, int n_in,
                              void* d_out, int out_size, void* d_ws, size_t ws_size,
                              hipStream_t stream) {
  const float* z = (const float*)d_in[0];         // [131072, 64]
  const float* C = (const float*)d_in[1];         // [512, 64]
  float* csq     = (float*)d_ws;                  // [512]
  float* partial = csq + KCL;                     // [8192]
  float* q       = (float*)d_out;                 // [131072, 512]
  float* loss    = q + (out_size - 1);            // scalar, last element

  const int nblocks = B_ROWS / 16;                // 8192
  TMC_csq_kernel<<<(KCL + 255) / 256, 256, 0, stream>>>(C, csq);
  TMC_cluster_kernel<<<nblocks, 256, 0, stream>>>(z, C, csq, q, partial);
  TMC_loss_kernel<<<1, 256, 0, stream>>>(partial, loss, nblocks);
}